// QRCNNLayer_47364899340468
// MI455X (gfx1250) — compile-verified
//
#include <hip/hip_runtime.h>
#include <hip/hip_bf16.h>

// CDNA5 (gfx1250) fp32 WMMA: D(16x16 f32) = A(16x4 f32) * B(4x16 f32) + C
typedef float v2f __attribute__((ext_vector_type(2)));
typedef float v8f __attribute__((ext_vector_type(8)));

__device__ __forceinline__ v8f wmma_f32(v2f a, v2f b, v8f c) {
  return __builtin_amdgcn_wmma_f32_16x16x4_f32(false, a, false, b, (short)0, c,
                                               false, false);
}

// ---------------------------------------------------------------------------
// Kernel 1: Conv1D (SAME, stride 1, NWC x WIO) as tiled WMMA GEMM.
//   y[b,t,f] = bias[f] + sum_{k,c} x[b,t+k-1,c] * W[k,c,f]
// GEMM view: M = B*T, N = F, Kdim = 3*C = 384. Output tile 128x128 per WG.
// A tile row-major (K contiguous); B tile staged TRANSPOSED (b_sT[f][k]) so
// every WMMA fragment is one contiguous ds_load_b64 (no operand movs).
// ---------------------------------------------------------------------------
#define AST 36   // A-tile stride: 128 rows x 32 k, padded (conflict-free)
#define BTS 36   // transposed B-tile stride: 128 f x 32 k, padded

__global__ void __launch_bounds__(256)
conv1d_wmma_f32(const float* __restrict__ x, const float* __restrict__ w,
                const float* __restrict__ bias, float* __restrict__ y) {
  const int T = 512, C = 128, F = 128;
  const int wg = blockIdx.x;
  const int b = wg >> 2;            // 4 row-tiles per batch (512/128)
  const int t0 = (wg & 3) * 128;

  __shared__ float a_s[128 * AST];   // 18 KB
  __shared__ float b_sT[128 * BTS];  // 18 KB (transposed W chunk)

  const int tid = threadIdx.x;
  const int lane = tid & 31;
  const int wv = tid >> 5;          // wave 0..7 -> M-tile (16 rows)
  const int m0 = wv * 16;
  const int half = lane >> 4;       // lane half selects K pair {0,1} vs {2,3}
  const int ln = lane & 15;
  const int koff = half * 2;

  v8f acc[8] = {};                  // 8 N-tiles of 16 cols each

  for (int kb = 0; kb < 384; kb += 32) {
    const int k = kb >> 7;          // conv tap 0..2
    const int c0 = kb & 127;        // channel base within tap
    __syncthreads();                // previous chunk's WMMA reads done

    if (kb + 32 < 384)              // warm L2/WGP$ for next W chunk
      __builtin_prefetch(w + (size_t)(kb + 32) * F + tid * 4, 0, 1);

    // Stage A chunk: a_s[r][j] = x[b, t0+r+k-1, c0+j], zero-padded in t.
    for (int i = 0; i < 4; ++i) {
      int q = tid + i * 256;        // 1024 float4 total
      int r = q >> 3;               // 0..127
      int j4 = (q & 7) * 4;         // 0..28
      int t = t0 + r + k - 1;
      float4 v = make_float4(0.f, 0.f, 0.f, 0.f);
      if ((unsigned)t < (unsigned)T)
        v = *(const float4*)(x + (((size_t)b * T + t) * C + c0 + j4));
      *(float4*)(&a_s[r * AST + j4]) = v;
    }
    // Stage B chunk TRANSPOSED: b_sT[f][kr] = W[kb+kr][f].
    // 4 coalesced global reads (consecutive f across lanes) per float4 store.
    for (int i = 0; i < 4; ++i) {
      int q = tid + i * 256;        // 1024 -> f in [0,128), k4 in {0,4,..,28}
      int f = q & 127;
      int k4 = (q >> 7) * 4;
      float4 v;
      v.x = w[(size_t)(kb + k4 + 0) * F + f];
      v.y = w[(size_t)(kb + k4 + 1) * F + f];
      v.z = w[(size_t)(kb + k4 + 2) * F + f];
      v.w = w[(size_t)(kb + k4 + 3) * F + f];
      *(float4*)(&b_sT[f * BTS + k4]) = v;   // row base 144B: 16B-aligned
    }
    __syncthreads();

#pragma unroll
    for (int ks = 0; ks < 8; ++ks) {
      const int kk = ks * 4 + koff;
      const v2f a = *(const v2f*)(&a_s[(m0 + ln) * AST + kk]);  // ds_load_b64
#pragma unroll
      for (int nt = 0; nt < 8; ++nt) {
        const v2f bb = *(const v2f*)(&b_sT[(nt * 16 + ln) * BTS + kk]);
        acc[nt] = wmma_f32(a, bb, acc[nt]);
      }
    }
  }

  // Bias + writeback. C/D layout: VGPR i -> row (i + half*8), col ln.
  for (int nt = 0; nt < 8; ++nt) {
    float bv = bias[nt * 16 + ln];
#pragma unroll
    for (int i = 0; i < 8; ++i) {
      int row = m0 + i + half * 8;
      y[(((size_t)b * T) + t0 + row) * F + nt * 16 + ln] = acc[nt][i] + bv;
    }
  }
}

// ---------------------------------------------------------------------------
// Kernel 2: batched QR (512x128 -> Q), in place over d_out.
// Whole matrix resident in LDS (512 x 132 floats = 264 KB of the 320 KB WGP
// LDS). Blocked classical Gram-Schmidt over 16-col panels; panel
// orthonormalization via CholeskyQR. All GEMMs use v_wmma_f32_16x16x4_f32.
// ---------------------------------------------------------------------------
#define QST 132  // LDS row stride for the resident matrix

__global__ void __launch_bounds__(256)
qr_cholqr_wmma(float* __restrict__ yq) {
  extern __shared__ float As[];          // 512 * QST floats (dynamic)
  __shared__ float Ss[7 * 16 * 16];      // -S (projection coefficients)
  __shared__ float Gp[8 * 16 * 16];      // per-wave partial Gram
  __shared__ float Gm[16 * 16];          // reduced Gram -> Cholesky R
  __shared__ float Ri[16 * 16];          // R^{-1}

  const int T = 512, F = 128;
  const int b = blockIdx.x;
  const int tid = threadIdx.x;
  const int lane = tid & 31, wv = tid >> 5;
  const int half = lane >> 4, ln = lane & 15, koff = half * 2;
  const size_t base = (size_t)b * T * F;

  // Load y[b] into LDS (coalesced float4).
  for (int i = 0; i < 64; ++i) {
    int q = tid + i * 256;
    int r = q >> 5;
    int c4 = (q & 31) * 4;
    *(float4*)(&As[r * QST + c4]) = *(const float4*)(yq + base + (size_t)r * F + c4);
  }
  __syncthreads();

  for (int p = 0; p < 8; ++p) {
    const int c0 = p * 16;

    // ---- block-CGS projection: V -= Q_prev (Q_prevᵀ V) ----
    if (p > 0) {
      if (wv < p) {                      // wave-uniform; EXEC stays full
        const int mcol = wv * 16;        // which 16 Q columns (M-tile)
        v8f s = {};
        for (int ks = 0; ks < 128; ++ks) {   // K = 512 rows
          const int kk = ks * 4 + koff;
          v2f a, bb;                     // column reads -> ds_load_2addr
          a.x = As[kk * QST + mcol + ln];        // A[m][k] = Q[k][m]
          a.y = As[(kk + 1) * QST + mcol + ln];
          bb.x = As[kk * QST + c0 + ln];         // B[k][j] = V[k][j]
          bb.y = As[(kk + 1) * QST + c0 + ln];
          s = wmma_f32(a, bb, s);
        }
#pragma unroll
        for (int i = 0; i < 8; ++i)      // store -S so update is pure FMA
          Ss[(wv * 16 + i + half * 8) * 16 + ln] = -s[i];
      }
      __syncthreads();

      // V += Q * (-S): M = 512 (32 row-tiles / 8 waves), K = 16p.
      for (int mt = wv; mt < 32; mt += 8) {
        const int r0 = mt * 16;
        v8f c;
#pragma unroll
        for (int i = 0; i < 8; ++i)
          c[i] = As[(r0 + i + half * 8) * QST + c0 + ln];
        for (int ks = 0; ks < p * 4; ++ks) {
          const int kk = ks * 4 + koff;
          const v2f a = *(const v2f*)(&As[(r0 + ln) * QST + kk]);  // b64
          v2f bb;
          bb.x = Ss[kk * 16 + ln];
          bb.y = Ss[(kk + 1) * 16 + ln];
          c = wmma_f32(a, bb, c);
        }
#pragma unroll
        for (int i = 0; i < 8; ++i)
          As[(r0 + i + half * 8) * QST + c0 + ln] = c[i];
      }
      __syncthreads();
    }

    // ---- CholeskyQR of panel: G = VᵀV (K=512 split over 8 waves) ----
    {
      v8f g = {};
      const int kbase = wv * 64;
      for (int ks = 0; ks < 16; ++ks) {
        const int kk = kbase + ks * 4 + koff;
        v2f f;                           // A and B fragments are identical
        f.x = As[kk * QST + c0 + ln];
        f.y = As[(kk + 1) * QST + c0 + ln];
        g = wmma_f32(f, f, g);
      }
#pragma unroll
      for (int i = 0; i < 8; ++i)
        Gp[wv * 256 + (i + half * 8) * 16 + ln] = g[i];
    }
    __syncthreads();
    {                                    // reduce 8 partials (one elem/thread)
      float a = 0.f;
#pragma unroll
      for (int w2 = 0; w2 < 8; ++w2) a += Gp[w2 * 256 + tid];
      Gm[tid] = a;
    }
    __syncthreads();

    if (tid == 0) {                      // 16x16 Cholesky + triangular invert
      for (int j = 0; j < 16; ++j) {
        float d = Gm[j * 16 + j];
        for (int t2 = 0; t2 < j; ++t2) { float v = Gm[t2 * 16 + j]; d -= v * v; }
        d = sqrtf(fmaxf(d, 1e-20f));
        Gm[j * 16 + j] = d;
        float inv = 1.f / d;
        for (int i2 = j + 1; i2 < 16; ++i2) {
          float s2 = Gm[j * 16 + i2];
          for (int t2 = 0; t2 < j; ++t2) s2 -= Gm[t2 * 16 + j] * Gm[t2 * 16 + i2];
          Gm[j * 16 + i2] = s2 * inv;    // R stored in upper triangle (row j)
        }
      }
      for (int i2 = 0; i2 < 256; ++i2) Ri[i2] = 0.f;
      for (int j = 15; j >= 0; --j) {
        Ri[j * 16 + j] = 1.f / Gm[j * 16 + j];
        for (int i2 = j - 1; i2 >= 0; --i2) {
          float s2 = 0.f;
          for (int t2 = i2 + 1; t2 <= j; ++t2) s2 += Gm[i2 * 16 + t2] * Ri[t2 * 16 + j];
          Ri[i2 * 16 + j] = -s2 / Gm[i2 * 16 + i2];
        }
      }
    }
    __syncthreads();

    // ---- V <- V * R^{-1}: M = 512, K = 16 ----
    for (int mt = wv; mt < 32; mt += 8) {
      const int r0 = mt * 16;
      v8f d = {};
#pragma unroll
      for (int ks = 0; ks < 4; ++ks) {
        const int kk = ks * 4 + koff;
        const v2f a = *(const v2f*)(&As[(r0 + ln) * QST + c0 + kk]);  // b64
        v2f bb;
        bb.x = Ri[kk * 16 + ln];
        bb.y = Ri[(kk + 1) * 16 + ln];
        d = wmma_f32(a, bb, d);
      }
#pragma unroll
      for (int i = 0; i < 8; ++i)
        As[(r0 + i + half * 8) * QST + c0 + ln] = d[i];
    }
    __syncthreads();
  }

  // Store Q back over d_out (coalesced float4).
  for (int i = 0; i < 64; ++i) {
    int q = tid + i * 256;
    int r = q >> 5;
    int c4 = (q & 31) * 4;
    *(float4*)(yq + base + (size_t)r * F + c4) = *(const float4*)(&As[r * QST + c4]);
  }
}

// ---------------------------------------------------------------------------
extern "C" void kernel_launch(void* const* d_in, const int* in_sizes, int n_in,
                              void* d_out, int out_size, void* d_ws, size_t ws_size,
                              hipStream_t stream) {
  (void)in_sizes; (void)n_in; (void)out_size; (void)d_ws; (void)ws_size;
  const float* x    = (const float*)d_in[0];   // [256,512,128] f32
  const float* w    = (const float*)d_in[1];   // [3,128,128]   f32
  const float* bias = (const float*)d_in[2];   // [128]         f32
  float* out = (float*)d_out;                  // [256,512,128] f32

  // Kernel 1: y = conv1d(x, W) + b, written into d_out (reused as scratch).
  conv1d_wmma_f32<<<dim3(256 * 4), dim3(256), 0, stream>>>(x, w, bias, out);

  // Kernel 2: per-batch QR in LDS, Q overwrites d_out. 512*132*4 B dynamic LDS.
  qr_cholqr_wmma<<<dim3(256), dim3(256), 512 * QST * sizeof(float), stream>>>(out);
}